// InferenceDynamicEmbeddingCollection_10728828305837
// MI455X (gfx1250) — compile-verified
//
#include <hip/hip_runtime.h>

// Embedding row gather (pooling=NONE):
//   out[i, 0:64] = table[values[i] % VOCAB, 0:64]
// N_VALUES = 4,194,304 rows, DIM = 64 fp32 (256 B/row), VOCAB = 1,000,000.
//
// Pure bandwidth kernel (0 FLOPs): ~1 GiB streamed out + ~1 GiB gathered in.
// MI455X strategy:
//   * B128 vector loads/stores (16B per lane; 16 lanes = one 256B row).
//   * NON-TEMPORAL stores for the 1 GiB output stream so it does not evict
//     the 256 MB table from the 192 MB L2 (each row is reused ~4.2x).
//   * Regular-temporal loads for the table so reused rows stay L2-resident.

typedef float v4f __attribute__((ext_vector_type(4)));

#define VOCAB_SZ 1000000u
#define CHUNKS_PER_ROW 16   // 64 floats / 4 per chunk

__global__ __launch_bounds__(256) void emb_gather_kernel(
    const int* __restrict__ values,     // [n_rows] int32 ids
    const v4f* __restrict__ table4,     // [VOCAB * 16] float4 view of table
    v4f* __restrict__ out4,             // [n_rows * 16] float4 view of output
    long long total_chunks)             // n_rows * 16
{
    long long tid = (long long)blockIdx.x * blockDim.x + threadIdx.x;
    if (tid >= total_chunks) return;

    long long row = tid >> 4;          // which embedding lookup
    int       c4  = (int)(tid & 15);   // which float4 within the 64-f32 row

    // 16 lanes share this load -> single broadcast transaction per row.
    unsigned idx = (unsigned)values[row] % VOCAB_SZ;

    // Regular-temporal B128 load: table rows are reused ~4x, keep in L2.
    v4f v = table4[(long long)idx * CHUNKS_PER_ROW + c4];

    // Non-temporal B128 store: 1 GiB write-once stream, bypass L2 retention
    // so it does not displace the embedding table.
    __builtin_nontemporal_store(v, out4 + tid);
}

extern "C" void kernel_launch(void* const* d_in, const int* in_sizes, int n_in,
                              void* d_out, int out_size, void* d_ws, size_t ws_size,
                              hipStream_t stream) {
    // setup_inputs() order: values (int), offsets (int, UNUSED), table (float32)
    const int*   values = (const int*)d_in[0];
    const float* table  = (const float*)d_in[2];
    float*       out    = (float*)d_out;

    const long long n_rows       = (long long)in_sizes[0];       // 4,194,304
    const long long total_chunks = n_rows * CHUNKS_PER_ROW;      // 67,108,864

    const int block = 256;  // 8 wave32 per block
    const long long grid = (total_chunks + block - 1) / block;   // 262,144

    emb_gather_kernel<<<dim3((unsigned)grid), dim3(block), 0, stream>>>(
        values, (const v4f*)table, (v4f*)out, total_chunks);
}